// LDPCBPDecoder_20925080666521
// MI455X (gfx1250) — compile-verified
//
#include <hip/hip_runtime.h>

// ---------------------------------------------------------------------------
// LDPC BP decoder for MI455X (gfx1250, wave32).
// State layout: msg[E][512] batch-contiguous -> every edge access is a fully
// coalesced 2KB row; lane = batch element, float4 (16B) per lane.
// Whole working set (~64MB) is L2-resident (192MB). CN-phase random gathers
// are staged through LDS with gfx1250 async global->LDS loads (ASYNCcnt).
// ---------------------------------------------------------------------------

#define N_VNS_C     8192
#define N_CNS_C     4096
#define NUM_EDGES_C 24576
#define BATCH_C     512
#define NUM_ITER_C  10

#define PHI_MIN_F   8.5e-8f
#define PHI_MAX_F   16.635532f
#define LLR_CLIP_F  20.0f

#if defined(__HIP_DEVICE_COMPILE__) && defined(__gfx1250__)
  #if __has_builtin(__builtin_amdgcn_global_load_async_to_lds_b128) && \
      __has_builtin(__builtin_amdgcn_s_wait_asynccnt)
    #define USE_ASYNC_COPY 1
    typedef int v4i_t __attribute__((vector_size(16)));
    typedef __attribute__((address_space(1))) v4i_t* g_v4i_ptr;
    typedef __attribute__((address_space(3))) v4i_t* l_v4i_ptr;
  #endif
#endif

// phi(x) = -log(tanh(x/2)) = log((1+e^-x)/(1-e^-x)), clipped like reference.
__device__ __forceinline__ float phi_f(float x) {
  x = fminf(fmaxf(x, PHI_MIN_F), PHI_MAX_F);
  float t = __expf(-x);
  return __logf(__fdividef(1.0f + t, 1.0f - t));
}

__device__ __forceinline__ float f4_get(const float4& v, int j) {
  return j == 0 ? v.x : (j == 1 ? v.y : (j == 2 ? v.z : v.w));
}
__device__ __forceinline__ void f4_set(float4& v, int j, float x) {
  if (j == 0) v.x = x; else if (j == 1) v.y = x; else if (j == 2) v.z = x; else v.w = x;
}

// dst[n*M + m] = -(clamp?)(src[m*N + n]); 32x32 LDS tile transpose.
__global__ __launch_bounds__(256) void transpose_neg(const float* __restrict__ src,
                                                     float* __restrict__ dst,
                                                     int M, int N, int do_clamp) {
  __shared__ float tile[32][33];
  const int bx = blockIdx.x;  // over N
  const int by = blockIdx.y;  // over M
  const int x = bx * 32 + threadIdx.x;
  #pragma unroll
  for (int i = threadIdx.y; i < 32; i += 8) {
    int y = by * 32 + i;
    float v = src[(size_t)y * N + x];
    if (do_clamp) v = fminf(fmaxf(v, -LLR_CLIP_F), LLR_CLIP_F);
    tile[i][threadIdx.x] = -v;
  }
  __syncthreads();
  const int ox = by * 32 + threadIdx.x;  // m
  #pragma unroll
  for (int i = threadIdx.y; i < 32; i += 8) {
    int oy = bx * 32 + i;                // n
    dst[(size_t)oy * M + ox] = tile[threadIdx.x][i];
  }
}

__global__ void zero_counts(int* __restrict__ counts) {
  int i = blockIdx.x * blockDim.x + threadIdx.x;
  if (i < N_CNS_C) counts[i] = 0;
}

// Inverse map: for each CN list its 6 edge ids (atomic fill; canonicalized by sort).
__global__ void cn_fill(const int* __restrict__ cn_con, int* __restrict__ cn_edges,
                        int* __restrict__ counts) {
  int e = blockIdx.x * blockDim.x + threadIdx.x;
  if (e < NUM_EDGES_C) {
    int c = cn_con[e];
    int p = atomicAdd(&counts[c], 1);
    cn_edges[c * 6 + p] = e;
  }
}

// Sort each CN's 6 edge ids ascending -> deterministic float summation order.
__global__ void cn_sort(int* __restrict__ cn_edges) {
  int c = blockIdx.x * blockDim.x + threadIdx.x;
  if (c >= N_CNS_C) return;
  int a[6];
  #pragma unroll
  for (int k = 0; k < 6; ++k) a[k] = cn_edges[c * 6 + k];
  #pragma unroll
  for (int i = 1; i < 6; ++i) {
    int key = a[i];
    int j = i - 1;
    while (j >= 0 && a[j] > key) { a[j + 1] = a[j]; --j; }
    a[j + 1] = key;
  }
  #pragma unroll
  for (int k = 0; k < 6; ++k) cn_edges[c * 6 + k] = a[k];
}

// VN phase, in place: edges of VN v are rows 3v..3v+2 (contiguous).
// Block = 256 threads = 2 VNs x 128 batch-quads (float4 per thread).
__global__ __launch_bounds__(256) void vn_update(float* __restrict__ msg,
                                                 const float* __restrict__ llr,
                                                 int first) {
  const int q = threadIdx.x & 127;
  const int v = blockIdx.x * 2 + (threadIdx.x >> 7);
  const int bo = q << 2;
  const float4 L = *(const float4*)(llr + (size_t)v * BATCH_C + bo);
  float* r0 = msg + (size_t)(3 * v) * BATCH_C + bo;
  float* r1 = r0 + BATCH_C;
  float* r2 = r1 + BATCH_C;
  float4 m0, m1, m2;
  if (first) {
    m0 = make_float4(0.f, 0.f, 0.f, 0.f); m1 = m0; m2 = m0;
  } else {
    m0 = *(const float4*)r0; m1 = *(const float4*)r1; m2 = *(const float4*)r2;
  }
  float4 t;
  t.x = L.x + m0.x + m1.x + m2.x;
  t.y = L.y + m0.y + m1.y + m2.y;
  t.z = L.z + m0.z + m1.z + m2.z;
  t.w = L.w + m0.w + m1.w + m2.w;
  float4 o0, o1, o2;
  o0.x = t.x - m0.x; o0.y = t.y - m0.y; o0.z = t.z - m0.z; o0.w = t.w - m0.w;
  o1.x = t.x - m1.x; o1.y = t.y - m1.y; o1.z = t.z - m1.z; o1.w = t.w - m1.w;
  o2.x = t.x - m2.x; o2.y = t.y - m2.y; o2.z = t.z - m2.z; o2.w = t.w - m2.w;
  *(float4*)r0 = o0; *(float4*)r1 = o1; *(float4*)r2 = o2;
}

// CN phase, in place. Block = 256 threads = 2 CNs x 128 batch-quads.
// Gathers 6 random 16B rows per lane via async global->LDS (ASYNCcnt), then
// boxplus-phi with sign-bit XOR parity.
__global__ __launch_bounds__(256) void cn_update(float* __restrict__ msg,
                                                 const int* __restrict__ cn_edges) {
  const int q = threadIdx.x & 127;
  const int s = threadIdx.x >> 7;
  const int c = blockIdx.x * 2 + s;
  const int bo = q << 2;
  int eidx[6];
  #pragma unroll
  for (int k = 0; k < 6; ++k) eidx[k] = cn_edges[c * 6 + k];

  float4 m[6];
#ifdef USE_ASYNC_COPY
  __shared__ float4 stage[2][6][128];
  #pragma unroll
  for (int k = 0; k < 6; ++k) {
    __builtin_amdgcn_global_load_async_to_lds_b128(
        (g_v4i_ptr)(msg + (size_t)eidx[k] * BATCH_C + bo),
        (l_v4i_ptr)&stage[s][k][q], 0, 0);
  }
  __builtin_amdgcn_s_wait_asynccnt(0);  // this wave's async ops landed in LDS
  #pragma unroll
  for (int k = 0; k < 6; ++k) m[k] = stage[s][k][q];
#else
  #pragma unroll
  for (int k = 0; k < 6; ++k)
    m[k] = *(const float4*)(msg + (size_t)eidx[k] * BATCH_C + bo);
#endif

  float4 o[6];
  #pragma unroll
  for (int j = 0; j < 4; ++j) {
    float p[6];
    unsigned sb[6];
    float S = 0.f;
    unsigned X = 0u;
    #pragma unroll
    for (int k = 0; k < 6; ++k) {
      float mv = f4_get(m[k], j);
      sb[k] = __float_as_uint(mv) & 0x80000000u;
      p[k] = phi_f(fabsf(mv));
      S += p[k];
      X ^= sb[k];
    }
    #pragma unroll
    for (int k = 0; k < 6; ++k) {
      float mag = phi_f(S - p[k]);  // phi out >= 0, sign bit clear
      f4_set(o[k], j, __uint_as_float(__float_as_uint(mag) | (X ^ sb[k])));
    }
  }
  #pragma unroll
  for (int k = 0; k < 6; ++k)
    *(float4*)(msg + (size_t)eidx[k] * BATCH_C + bo) = o[k];
}

// x_tot accumulated in place into llr buffer.
__global__ __launch_bounds__(256) void final_tot(const float* __restrict__ msg,
                                                 float* __restrict__ llr) {
  const int q = threadIdx.x & 127;
  const int v = blockIdx.x * 2 + (threadIdx.x >> 7);
  const int bo = q << 2;
  float* Lp = llr + (size_t)v * BATCH_C + bo;
  const float* r0 = msg + (size_t)(3 * v) * BATCH_C + bo;
  float4 L = *(const float4*)Lp;
  float4 m0 = *(const float4*)r0;
  float4 m1 = *(const float4*)(r0 + BATCH_C);
  float4 m2 = *(const float4*)(r0 + 2 * BATCH_C);
  L.x += m0.x + m1.x + m2.x;
  L.y += m0.y + m1.y + m2.y;
  L.z += m0.z + m1.z + m2.z;
  L.w += m0.w + m1.w + m2.w;
  *(float4*)Lp = L;
}

extern "C" void kernel_launch(void* const* d_in, const int* in_sizes, int n_in,
                              void* d_out, int out_size, void* d_ws, size_t ws_size,
                              hipStream_t stream) {
  (void)in_sizes; (void)n_in; (void)out_size; (void)ws_size;
  const float* llrs_ch = (const float*)d_in[0];
  // d_in[1] = vn_con: implicit (edges are VN-major, vn_con[e] == e/3)
  const int* cn_con = (const int*)d_in[2];
  float* out = (float*)d_out;

  // Workspace layout (~67.2 MB):
  float* msg      = (float*)d_ws;                           // [E][512]   48 MB
  float* llr      = msg + (size_t)NUM_EDGES_C * BATCH_C;    // [V][512]   16 MB
  int*   cn_edges = (int*)(llr + (size_t)N_VNS_C * BATCH_C);// [C][6]     96 KB
  int*   counts   = cn_edges + NUM_EDGES_C;                 // [C]        16 KB

  // llr_int[v][b] = -clamp(llrs_ch[b][v], +-20)
  transpose_neg<<<dim3(N_VNS_C / 32, BATCH_C / 32), dim3(32, 8), 0, stream>>>(
      llrs_ch, llr, BATCH_C, N_VNS_C, 1);

  // Deterministic CN->edge map
  zero_counts<<<N_CNS_C / 256, 256, 0, stream>>>(counts);
  cn_fill<<<NUM_EDGES_C / 256, 256, 0, stream>>>(cn_con, cn_edges, counts);
  cn_sort<<<N_CNS_C / 256, 256, 0, stream>>>(cn_edges);

  for (int it = 0; it < NUM_ITER_C; ++it) {
    vn_update<<<N_VNS_C / 2, 256, 0, stream>>>(msg, llr, it == 0 ? 1 : 0);
    cn_update<<<N_CNS_C / 2, 256, 0, stream>>>(msg, cn_edges);
  }

  final_tot<<<N_VNS_C / 2, 256, 0, stream>>>(msg, llr);

  // out[b][v] = -x_tot[v][b]
  transpose_neg<<<dim3(BATCH_C / 32, N_VNS_C / 32), dim3(32, 8), 0, stream>>>(
      llr, out, N_VNS_C, BATCH_C, 0);
}